// NMS_10222022165053
// MI455X (gfx1250) — compile-verified
//
#include <hip/hip_runtime.h>
#include <stdint.h>

// Problem constants (from reference)
#define NIMG      16
#define NBOX      25200
#define NCLS      80
#define CSTRIDE   85          // 4 box + 1 obj + 80 cls
#define KCAND     1024
#define MAXDET    1000
#define CONF_T    0.25f
#define IOU_T     0.45f

// Kernel-1 tiling
#define TILE_ROWS  128
#define K1_THREADS 128
#define TPI ((NBOX + TILE_ROWS - 1) / TILE_ROWS)   // tiles per image = 197

typedef __attribute__((ext_vector_type(4))) unsigned int uint4v;
typedef __attribute__((ext_vector_type(8))) int          int8v;
typedef __attribute__((ext_vector_type(4))) int          int4v;
typedef __attribute__((ext_vector_type(2))) float        v2f;
typedef __attribute__((ext_vector_type(8))) float        v8f;

// ---------------------------------------------------------------------------
// Kernel 1: stream pred through LDS via the Tensor Data Mover; per box compute
// conf = obj * max(cls) and store masked conf (HBM-bound stage, ~137MB read).
// ---------------------------------------------------------------------------
__global__ __launch_bounds__(K1_THREADS) void nms_score_kernel(
    const float* __restrict__ pred, float* __restrict__ conf_ws) {
  __shared__ float tile[TILE_ROWS * CSTRIDE];

  const int tid    = threadIdx.x;
  const int tileId = blockIdx.x;
  const int img    = tileId / TPI;
  const int t      = tileId % TPI;
  const int row0   = t * TILE_ROWS;
  int rows = NBOX - row0;
  if (rows > TILE_ROWS) rows = TILE_ROWS;

  const float* src = pred + ((size_t)img * NBOX + row0) * CSTRIDE;

#if __has_builtin(__builtin_amdgcn_tensor_load_to_lds) && __has_builtin(__builtin_amdgcn_s_wait_tensorcnt)
  // One wave issues the TDM DMA of rows x 85 f32 into LDS (TENSORcnt-tracked).
  if ((tid >> 5) == 0) {
    unsigned long long ga = (unsigned long long)(uintptr_t)src;
    unsigned int lds_off  = (unsigned int)(uintptr_t)(&tile[0]); // low 32 bits = LDS byte addr

    uint4v g0;
    g0[0] = 1u;                                   // count=1, user-mode, no gather
    g0[1] = lds_off;                              // lds_addr
    g0[2] = (unsigned int)(ga & 0xFFFFFFFFull);   // global_addr[31:0]
    g0[3] = (unsigned int)((ga >> 32) & 0x1FFFFFFull) | (2u << 30); // addr[56:32] | type=2

    int8v g1;
    g1[0] = (int)(2u << 16);                               // data_size = 4 bytes
    g1[1] = (int)(((unsigned)CSTRIDE & 0xFFFFu) << 16);    // tensor_dim0[15:0] = 85
    g1[2] = (int)((((unsigned)rows) & 0xFFFFu) << 16);     // dim0 hi=0 | tensor_dim1 lo = rows
    g1[3] = (int)(((unsigned)CSTRIDE & 0xFFFFu) << 16);    // dim1 hi=0 | tile_dim0 = 85
    g1[4] = (int)(((unsigned)rows) & 0xFFFFu);             // tile_dim1 = rows, tile_dim2 = 0
    g1[5] = CSTRIDE;                                       // tensor_dim0_stride = 85
    g1[6] = 0;
    g1[7] = 0;

    int4v g2 = {0, 0, 0, 0};
    int4v g3 = {0, 0, 0, 0};
#if __clang_major__ >= 23
    int8v g4 = {0, 0, 0, 0, 0, 0, 0, 0};
    __builtin_amdgcn_tensor_load_to_lds(g0, g1, g2, g3, g4, 0);
#else
    __builtin_amdgcn_tensor_load_to_lds(g0, g1, g2, g3, 0);
#endif
    __builtin_amdgcn_s_wait_tensorcnt(0);
  }
#else
  for (int i = tid; i < rows * CSTRIDE; i += K1_THREADS) tile[i] = src[i];
#endif
  __syncthreads();

  if (tid < rows) {
    const float* r = &tile[tid * CSTRIDE];
    float m = r[5];
#pragma unroll 4
    for (int c = 1; c < NCLS; ++c) {
      float v = r[5 + c];
      m = (v > m) ? v : m;
    }
    float conf = r[4] * m;  // obj >= 0 -> max(obj*cls) = obj*max(cls)
    conf_ws[(size_t)img * NBOX + row0 + tid] = (conf > CONF_T) ? conf : -1.0f;
  }
}

// ---------------------------------------------------------------------------
// Kernel 2: one 1024-thread block per image.
//   radix-select exact top-1024 -> bitonic sort (desc conf, asc idx)
//   -> WMMA xywh->xyxy transform -> tiled greedy NMS scan -> compacted output.
// ---------------------------------------------------------------------------
__global__ __launch_bounds__(1024) void nms_select_kernel(
    const float* __restrict__ pred, const float* __restrict__ conf_ws,
    float* __restrict__ dout) {
  __shared__ int          s_hist[256];
  __shared__ unsigned int s_pivot;
  __shared__ int          s_kr;
  __shared__ int          s_cnt;
  __shared__ float        s_conf[KCAND];
  __shared__ int          s_gidx[KCAND];
  __shared__ float        s_x1[KCAND], s_y1[KCAND], s_x2[KCAND], s_y2[KCAND];
  __shared__ float        s_area[KCAND];
  __shared__ int          s_cls[KCAND];
  __shared__ unsigned int s_tileMask;
  __shared__ int          s_wsum[32];

  const int tid  = threadIdx.x;
  const int lane = tid & 31;
  const int wv   = tid >> 5;
  const int img  = blockIdx.x;
  const float* confb = conf_ws + (size_t)img * NBOX;

  // ---- exact top-K threshold via 4-pass MSB-first radix select on float bits
  unsigned int pivot = 0;
  int kr = KCAND;
  for (int shift = 24; shift >= 0; shift -= 8) {
    for (int i = tid; i < 256; i += 1024) s_hist[i] = 0;
    __syncthreads();
    const unsigned int pmask = (shift == 24) ? 0u : (0xFFFFFFFFu << (shift + 8));
    for (int i = tid; i < NBOX; i += 1024) {
      float c = confb[i];
      unsigned int key = (c > 0.0f) ? __float_as_uint(c) : 0u;  // positives order as uints
      if ((key & pmask) == pivot) atomicAdd(&s_hist[(key >> shift) & 255], 1);
    }
    __syncthreads();
    if (tid == 0) {
      int acc = 0, d = 255;
      for (; d > 0; --d) {
        int c = s_hist[d];
        if (acc + c >= kr) break;
        acc += c;
      }
      s_pivot = pivot | ((unsigned int)d << shift);
      s_kr = kr - acc;
    }
    __syncthreads();
    pivot = s_pivot;
    kr = s_kr;
  }

  // ---- compact: all keys > pivot (< K of them), then fill with == pivot
  if (tid == 0) s_cnt = 0;
  __syncthreads();
  for (int i = tid; i < NBOX; i += 1024) {
    float c = confb[i];
    unsigned int key = (c > 0.0f) ? __float_as_uint(c) : 0u;
    if (key > pivot) {
      int s = atomicAdd(&s_cnt, 1);
      s_conf[s] = c;
      s_gidx[s] = i;
    }
  }
  __syncthreads();
  for (int i = tid; i < NBOX; i += 1024) {
    float c = confb[i];
    unsigned int key = (c > 0.0f) ? __float_as_uint(c) : 0u;
    if (key == pivot) {
      int s = atomicAdd(&s_cnt, 1);
      if (s < KCAND) {
        s_conf[s] = c;
        s_gidx[s] = i;
      }
    }
  }
  __syncthreads();

  // ---- bitonic sort of 1024 (conf desc, gidx asc on ties -> matches top_k)
  for (int size = 2; size <= KCAND; size <<= 1) {
    for (int stride = size >> 1; stride > 0; stride >>= 1) {
      __syncthreads();
      int i = tid, j = i ^ stride;
      if (j > i) {
        float ci = s_conf[i], cj = s_conf[j];
        int   gi = s_gidx[i], gj = s_gidx[j];
        bool jBeats = (cj > ci) || (cj == ci && gj < gi);
        bool iBeats = (ci > cj) || (ci == cj && gi < gj);
        bool doSwap = ((i & size) == 0) ? jBeats : iBeats;
        if (doSwap) {
          s_conf[i] = cj; s_conf[j] = ci;
          s_gidx[i] = gj; s_gidx[j] = gi;
        }
      }
    }
  }
  __syncthreads();

  // ---- per-candidate class argmax (thread tid -> candidate tid)
  {
    float c = s_conf[tid];
    int cl = 0;
    if (c > 0.0f) {
      const float* r = pred + ((size_t)img * NBOX + s_gidx[tid]) * CSTRIDE;
      float m = r[5];
      for (int cc = 1; cc < NCLS; ++cc) {
        float v = r[5 + cc];
        if (v > m) { m = v; cl = cc; }
      }
    }
    s_cls[tid] = cl;
  }

  // ---- xywh -> xyxy as D = A(16x4) x B(4x16) on the matrix pipe.
  // A = [[1,0,-.5,0],[0,1,0,-.5],[1,0,.5,0],[0,1,0,.5]] padded to 16 rows.
  // Exact-zero / power-of-two coefficients => bit-identical to scalar path.
#if __has_builtin(__builtin_amdgcn_wmma_f32_16x16x4_f32)
  {
    const int  m16 = lane & 15;
    const bool hi  = lane >= 16;   // holds K=2,3 halves per ISA 16x4 f32 layout
    v2f a;
    a[0] = hi ? ((m16 == 0) ? -0.5f : (m16 == 2) ? 0.5f : 0.0f)   // K=2 (w)
              : ((m16 == 0 || m16 == 2) ? 1.0f : 0.0f);           // K=0 (cx)
    a[1] = hi ? ((m16 == 1) ? -0.5f : (m16 == 3) ? 0.5f : 0.0f)   // K=3 (h)
              : ((m16 == 1 || m16 == 3) ? 1.0f : 0.0f);           // K=1 (cy)
#pragma unroll
    for (int g = 0; g < 2; ++g) {
      int cand = (wv << 5) + (g << 4) + m16;
      const float* r = pred + ((size_t)img * NBOX + s_gidx[cand]) * CSTRIDE;
      bool valid = s_conf[cand] > 0.0f;
      int off = hi ? 2 : 0;                       // lanes 0-15: cx,cy ; 16-31: w,h
      v2f b;
      b[0] = valid ? r[off]     : 0.0f;
      b[1] = valid ? r[off + 1] : 0.0f;
      v8f c = {};
      v8f d = __builtin_amdgcn_wmma_f32_16x16x4_f32(false, a, false, b,
                                                    (short)0, c, false, false);
      if (!hi) {                                  // D rows 0-3 live in vgpr0-3, lanes 0-15
        s_x1[cand] = d[0]; s_y1[cand] = d[1];
        s_x2[cand] = d[2]; s_y2[cand] = d[3];
        s_area[cand] = (d[2] - d[0]) * (d[3] - d[1]);
      }
    }
  }
#else
  {
    float c = s_conf[tid];
    float x1 = 0.f, y1 = 0.f, x2 = 0.f, y2 = 0.f;
    if (c > 0.0f) {
      const float* r = pred + ((size_t)img * NBOX + s_gidx[tid]) * CSTRIDE;
      float cx = r[0], cy = r[1], w = r[2], h = r[3];
      x1 = cx - w * 0.5f; y1 = cy - h * 0.5f;
      x2 = cx + w * 0.5f; y2 = cy + h * 0.5f;
    }
    s_x1[tid] = x1; s_y1[tid] = y1; s_x2[tid] = x2; s_y2[tid] = y2;
    s_area[tid] = (x2 - x1) * (y2 - y1);
  }
#endif
  __syncthreads();

  const float mx1 = s_x1[tid], my1 = s_y1[tid], mx2 = s_x2[tid], my2 = s_y2[tid];
  const float marea = s_area[tid];
  const int   mcls  = s_cls[tid];
  const float mconf = s_conf[tid];

  // ---- tiled greedy scan: wave t serially resolves its 32 boxes (shuffles),
  //      then whole group applies the tile's keep-mask (2 barriers / tile).
  bool sup = false, keep = false;
  for (int tI = 0; tI < KCAND / 32; ++tI) {
    if (wv == tI) {
      for (int k = 0; k < 32; ++k) {
        int jk = (tI << 5) + k;
        int sk = __shfl((int)sup, k, 32);                 // wave32 broadcast
        bool keep_k = (!sk) && (s_conf[jk] > 0.0f);
        if (keep_k) {
          if (!sup && s_cls[jk] == mcls) {                // class offset => cross-class IoU==0
            float ix1 = fmaxf(mx1, s_x1[jk]);
            float iy1 = fmaxf(my1, s_y1[jk]);
            float ix2 = fminf(mx2, s_x2[jk]);
            float iy2 = fminf(my2, s_y2[jk]);
            float inter = fmaxf(ix2 - ix1, 0.0f) * fmaxf(iy2 - iy1, 0.0f);
            float u = marea + s_area[jk] - inter + 1e-9f;
            if (inter / u > IOU_T) sup = true;
          }
          if (lane == k) sup = true;                      // self suppressed after keep
        }
        if (lane == k) keep = keep_k;
      }
      unsigned long long bm = __ballot(keep);             // 32-bit mask on wave32
      if (lane == 0) s_tileMask = (unsigned int)bm;
    }
    __syncthreads();
    unsigned int mask = s_tileMask;
    if (wv != tI && !sup && mconf > 0.0f) {
      while (mask) {
        int k = __ffs(mask) - 1;
        mask &= mask - 1;
        int jk = (tI << 5) + k;
        if (s_cls[jk] == mcls) {
          float ix1 = fmaxf(mx1, s_x1[jk]);
          float iy1 = fmaxf(my1, s_y1[jk]);
          float ix2 = fminf(mx2, s_x2[jk]);
          float iy2 = fminf(my2, s_y2[jk]);
          float inter = fmaxf(ix2 - ix1, 0.0f) * fmaxf(iy2 - iy1, 0.0f);
          float u = marea + s_area[jk] - inter + 1e-9f;
          if (inter / u > IOU_T) { sup = true; break; }
        }
      }
    }
    __syncthreads();
  }

  // ---- stable compaction of keep flags (candidates already sorted desc, so
  //      this IS top_k(sel_score, 1000)): ballot + popc prefix sum.
  unsigned int wmask = (unsigned int)__ballot(keep);
  int inwave = __popc(wmask & ((1u << lane) - 1u));
  if (lane == 0) s_wsum[wv] = __popc(wmask);
  __syncthreads();
  if (tid < 32) {
    int v = s_wsum[tid];
    int orig = v;
    for (int off = 1; off < 32; off <<= 1) {
      int n = __shfl_up(v, off, 32);
      if (tid >= off) v += n;
    }
    s_wsum[tid] = v - orig;  // exclusive scan
  }
  __syncthreads();
  int rank = s_wsum[wv] + inwave;

  float* outb = dout + (size_t)img * MAXDET * 6;
  for (int i = tid; i < MAXDET * 6; i += 1024) outb[i] = 0.0f;
  __syncthreads();
  if (keep && rank < MAXDET) {
    float* o = outb + rank * 6;
    o[0] = mx1; o[1] = my1; o[2] = mx2; o[3] = my2;
    o[4] = mconf; o[5] = (float)mcls;
  }
}

// ---------------------------------------------------------------------------
extern "C" void kernel_launch(void* const* d_in, const int* in_sizes, int n_in,
                              void* d_out, int out_size, void* d_ws, size_t ws_size,
                              hipStream_t stream) {
  (void)in_sizes; (void)n_in; (void)out_size; (void)ws_size;
  const float* pred = (const float*)d_in[0];
  float* conf_ws = (float*)d_ws;               // NIMG*NBOX floats = 1.6 MB
  float* out = (float*)d_out;                  // NIMG*1000*6 floats

  nms_score_kernel<<<NIMG * TPI, K1_THREADS, 0, stream>>>(pred, conf_ws);
  nms_select_kernel<<<NIMG, 1024, 0, stream>>>(pred, conf_ws, out);
}